// Consensus_549755813978
// MI455X (gfx1250) — compile-verified
//
#include <hip/hip_runtime.h>
#include <hip/hip_bf16.h>

typedef __attribute__((ext_vector_type(2))) float v2f;
typedef __attribute__((ext_vector_type(8))) float v8f;

#define TB 16
#define TC 768
#define THW 576
#define TN (TB * THW)                 // 9216
#define TTOT ((size_t)TB * TC * THW)  // 7,077,888

// ---------------------------------------------------------------------------
// K1: 1x1 conv GEMM:  Y[b,o,hw] = sum_c W[o,c] * X[b,c,hw] + bias[o] (+ X[b,o,hw])
// grid: (hw_tiles=9, o_tiles=6, b=16), block 128 (4 waves).
// Workgroup tile 128(o) x 64(hw); wave tile 32x64 = acc[2][4] for frag reuse.
// ---------------------------------------------------------------------------
template <bool RES>
__global__ __launch_bounds__(128) void k_conv1x1(const float* __restrict__ Wt,
                                                 const float* __restrict__ bias,
                                                 const float* __restrict__ X,
                                                 float* __restrict__ Y) {
  const int b  = blockIdx.z;
  const int o0 = blockIdx.y * 128;
  const int n0 = blockIdx.x * 64;
  __shared__ float Ws[128][20];   // stride 20 floats: rows 16B aligned, conflict-free
  __shared__ float Xs[16][68];    // stride 68 floats: rows 16B aligned
  const int tid  = threadIdx.x;
  const int wave = tid >> 5;
  const int lane = tid & 31;
  const int hi   = lane >> 4;
  const int l16  = lane & 15;
  const float* Xb = X + (size_t)b * (TC * THW);
  v8f acc[2][4] = {};

  for (int k0 = 0; k0 < TC; k0 += 16) {
    // Ws: 128x16 = 512 float4, 128 threads -> 4 each
    for (int idx = tid; idx < 512; idx += 128) {
      const int r = idx >> 2, c4 = (idx & 3) * 4;
      const float4 v = *(const float4*)&Wt[(size_t)(o0 + r) * TC + k0 + c4];
      *(float4*)&Ws[r][c4] = v;
    }
    // Xs: 16x64 = 256 float4, 128 threads -> 2 each
    for (int idx = tid; idx < 256; idx += 128) {
      const int r = idx >> 4, c4 = (idx & 15) * 4;
      const float4 v = *(const float4*)&Xb[(size_t)(k0 + r) * THW + n0 + c4];
      *(float4*)&Xs[r][c4] = v;
    }
    __syncthreads();
#pragma unroll
    for (int kk = 0; kk < 16; kk += 4) {
      const int ka = kk + (hi ? 2 : 0);
      v2f a[2];
#pragma unroll
      for (int t = 0; t < 2; ++t) {
        const int mrow = wave * 32 + t * 16 + l16;
        a[t].x = Ws[mrow][ka];
        a[t].y = Ws[mrow][ka + 1];
      }
      v2f bb[4];
#pragma unroll
      for (int j = 0; j < 4; ++j) {
        bb[j].x = Xs[ka][j * 16 + l16];
        bb[j].y = Xs[ka + 1][j * 16 + l16];
      }
#pragma unroll
      for (int t = 0; t < 2; ++t)
#pragma unroll
        for (int j = 0; j < 4; ++j)
          acc[t][j] = __builtin_amdgcn_wmma_f32_16x16x4_f32(
              false, a[t], false, bb[j], (short)0, acc[t][j], false, false);
    }
    __syncthreads();
  }
#pragma unroll
  for (int t = 0; t < 2; ++t)
#pragma unroll
    for (int j = 0; j < 4; ++j)
#pragma unroll
      for (int r = 0; r < 8; ++r) {
        const int row = o0 + wave * 32 + t * 16 + (hi ? r + 8 : r);
        const int col = n0 + j * 16 + l16;
        float v = acc[t][j][r] + bias[row];
        if (RES) v += Xb[(size_t)row * THW + col];
        Y[(size_t)b * (TC * THW) + (size_t)row * THW + col] = v;
      }
}

// ---------------------------------------------------------------------------
// K2: fused attention + max:  red2[n][b2] = scale * max_hw( q[n,:] . k[:, b2*HW+hw] )
// grid: (576 row-tiles, 16 b2), block 128 (4 waves).
// Each wave: 9 col-tiles as 3 groups of 3 concurrent tiles (A-frag reuse x3).
// ---------------------------------------------------------------------------
__global__ __launch_bounds__(128) void k_attn_max(const float* __restrict__ Q,
                                                  const float* __restrict__ K,
                                                  float* __restrict__ red2) {
  const int nt   = blockIdx.x;
  const int b2   = blockIdx.y;
  const int tid  = threadIdx.x;
  const int wave = tid >> 5;
  const int lane = tid & 31;
  const int hi   = lane >> 4;
  const int l16  = lane & 15;
  const int bq   = nt / 36;  // 36 row-tiles per batch
  const int hw0  = (nt % 36) * 16;
  __shared__ float Qs[16][68];        // stride 68: rows 16B aligned
  __shared__ float Ks[4][64][52];     // stride 52: rows 16B aligned
  __shared__ float wmax[4][16];
  const float* Qb = Q + (size_t)bq * (TC * THW) + hw0;
  const float* Kb = K + (size_t)b2 * (TC * THW);

  float rmax[8];
#pragma unroll
  for (int r = 0; r < 8; ++r) rmax[r] = -3.4e38f;

  for (int grp = 0; grp < 3; ++grp) {
    const int col0 = (wave * 9 + grp * 3) * 16;  // 48 cols per group
    v8f acc[3] = {};
    for (int kc = 0; kc < TC; kc += 64) {
      // Qs: 16 rows x 64 ch (all threads; r contiguous in memory)
      for (int idx = tid; idx < 16 * 64; idx += 128) {
        const int r = idx & 15, c = idx >> 4;
        Qs[r][c] = Qb[(size_t)(kc + c) * THW + r];
      }
      // per-wave Ks: 64 ch x 48 cols as float4 (768 float4 / 32 lanes)
      for (int idx = lane; idx < 64 * 12; idx += 32) {
        const int r = idx / 12, c4 = (idx % 12) * 4;
        const float4 v = *(const float4*)&Kb[(size_t)(kc + r) * THW + col0 + c4];
        *(float4*)&Ks[wave][r][c4] = v;
      }
      __syncthreads();
#pragma unroll
      for (int kk = 0; kk < 64; kk += 4) {
        const int ka = kk + (hi ? 2 : 0);
        v2f a;
        a.x = Qs[l16][ka];
        a.y = Qs[l16][ka + 1];
#pragma unroll
        for (int t = 0; t < 3; ++t) {
          v2f bb;
          bb.x = Ks[wave][ka][t * 16 + l16];
          bb.y = Ks[wave][ka + 1][t * 16 + l16];
          acc[t] = __builtin_amdgcn_wmma_f32_16x16x4_f32(
              false, a, false, bb, (short)0, acc[t], false, false);
        }
      }
      __syncthreads();
    }
#pragma unroll
    for (int t = 0; t < 3; ++t)
#pragma unroll
      for (int r = 0; r < 8; ++r) rmax[r] = fmaxf(rmax[r], acc[t][r]);
  }
  // max across the 16 lanes holding the same row (stays within 16-lane halves)
#pragma unroll
  for (int m = 1; m < 16; m <<= 1) {
#pragma unroll
    for (int r = 0; r < 8; ++r) rmax[r] = fmaxf(rmax[r], __shfl_xor(rmax[r], m, 32));
  }
  if (l16 == 0) {
#pragma unroll
    for (int r = 0; r < 8; ++r) wmax[wave][(hi ? 8 : 0) + r] = rmax[r];
  }
  __syncthreads();
  if (tid < 16) {
    float m = fmaxf(fmaxf(wmax[0][tid], wmax[1][tid]), fmaxf(wmax[2][tid], wmax[3][tid]));
    red2[(size_t)(nt * 16 + tid) * TB + b2] = m * 0.03608439182435161f;  // 1/sqrt(768)
  }
}

// ---------------------------------------------------------------------------
// inverse L2 norm over channels per pixel
// ---------------------------------------------------------------------------
__global__ __launch_bounds__(256) void k_invnorm(const float* __restrict__ X,
                                                 float* __restrict__ inorm) {
  const int n = blockIdx.x * 256 + threadIdx.x;
  if (n >= TN) return;
  const int b = n / THW, hw = n % THW;
  const float* p = X + (size_t)b * (TC * THW) + hw;
  float s = 0.f;
  for (int c = 0; c < TC; ++c) {
    float v = p[(size_t)c * THW];
    s += v * v;
  }
  inorm[n] = 1.0f / fmaxf(sqrtf(s), 1e-12f);
}

// ---------------------------------------------------------------------------
// mean over b2, softmax over hw, tie-mask at softmax max. grid 16, block 576
// ---------------------------------------------------------------------------
__global__ __launch_bounds__(576) void k_softmax_mask(const float* __restrict__ red2,
                                                      float* __restrict__ mask) {
  const int b = blockIdx.x;
  const int hw = threadIdx.x;
  const int n = b * THW + hw;
  float s = 0.f;
#pragma unroll
  for (int b2 = 0; b2 < TB; ++b2) s += red2[(size_t)n * TB + b2];
  s *= (1.0f / TB);
  __shared__ float red[576];
  // max
  red[hw] = s;
  __syncthreads();
  if (hw < 64) red[hw] = fmaxf(red[hw], red[hw + 512]);
  __syncthreads();
  for (int off = 256; off >= 1; off >>= 1) {
    if (hw < off) red[hw] = fmaxf(red[hw], red[hw + off]);
    __syncthreads();
  }
  const float mx = red[0];
  __syncthreads();
  const float e = __expf(s - mx);
  // sum
  red[hw] = e;
  __syncthreads();
  if (hw < 64) red[hw] += red[hw + 512];
  __syncthreads();
  for (int off = 256; off >= 1; off >>= 1) {
    if (hw < off) red[hw] += red[hw + off];
    __syncthreads();
  }
  const float sum = red[0];
  __syncthreads();
  const float soft = e / sum;
  // max of softmax
  red[hw] = soft;
  __syncthreads();
  if (hw < 64) red[hw] = fmaxf(red[hw], red[hw + 512]);
  __syncthreads();
  for (int off = 256; off >= 1; off >>= 1) {
    if (hw < off) red[hw] = fmaxf(red[hw], red[hw + off]);
    __syncthreads();
  }
  mask[n] = (soft == red[0]) ? 1.0f : 0.0f;
}

// ---------------------------------------------------------------------------
// seeds[b][c] = sum_hw mask * inorm * X   (grid: (3, 16), block 256)
// ---------------------------------------------------------------------------
__global__ __launch_bounds__(256) void k_seeds(const float* __restrict__ X,
                                               const float* __restrict__ inorm,
                                               const float* __restrict__ mask,
                                               float* __restrict__ seeds) {
  const int b = blockIdx.y;
  const int c = blockIdx.x * 256 + threadIdx.x;
  const float* p = X + (size_t)b * (TC * THW) + (size_t)c * THW;
  const float* mk = mask + b * THW;
  const float* in = inorm + b * THW;
  float s = 0.f;
  for (int hw = 0; hw < THW; ++hw) s += mk[hw] * in[hw] * p[hw];
  seeds[b * TC + c] = s;
}

// ---------------------------------------------------------------------------
// cor[n] = inorm[n]/16 * sum_o relu( X[b,:,hw] . seeds[o,:] )
// ---------------------------------------------------------------------------
__global__ __launch_bounds__(256) void k_cor(const float* __restrict__ X,
                                             const float* __restrict__ inorm,
                                             const float* __restrict__ seeds,
                                             float* __restrict__ cor) {
  const int n = blockIdx.x * 256 + threadIdx.x;
  if (n >= TN) return;
  const int b = n / THW, hw = n % THW;
  const float* p = X + (size_t)b * (TC * THW) + hw;
  float acc[TB] = {};
  for (int c = 0; c < TC; ++c) {
    const float v = p[(size_t)c * THW];
#pragma unroll
    for (int o = 0; o < TB; ++o) acc[o] = fmaf(v, seeds[o * TC + c], acc[o]);
  }
  float s = 0.f;
#pragma unroll
  for (int o = 0; o < TB; ++o) s += fmaxf(acc[o], 0.f);
  cor[n] = s * inorm[n] * (1.0f / TB);
}

// ---------------------------------------------------------------------------
// per-batch min/max of cor. grid 16, block 576
// ---------------------------------------------------------------------------
__global__ __launch_bounds__(576) void k_minmax(const float* __restrict__ cor,
                                                float* __restrict__ cmm) {
  const int b = blockIdx.x, hw = threadIdx.x;
  __shared__ float rmn[576], rmx[576];
  const float v = cor[b * THW + hw];
  rmn[hw] = v;
  rmx[hw] = v;
  __syncthreads();
  if (hw < 64) {
    rmn[hw] = fminf(rmn[hw], rmn[hw + 512]);
    rmx[hw] = fmaxf(rmx[hw], rmx[hw + 512]);
  }
  __syncthreads();
  for (int off = 256; off >= 1; off >>= 1) {
    if (hw < off) {
      rmn[hw] = fminf(rmn[hw], rmn[hw + off]);
      rmx[hw] = fmaxf(rmx[hw], rmx[hw + off]);
    }
    __syncthreads();
  }
  if (hw == 0) {
    cmm[b * 2] = rmn[0];
    cmm[b * 2 + 1] = rmx[0];
  }
}

// ---------------------------------------------------------------------------
// x_cur = (addprev ? x_cur : 0) + x_conv * cor_norm
// ---------------------------------------------------------------------------
__global__ __launch_bounds__(256) void k_scale_res(const float* __restrict__ Xc,
                                                   const float* __restrict__ cor,
                                                   const float* __restrict__ cmm,
                                                   float* __restrict__ Xcur,
                                                   int addprev) {
  const size_t i = (size_t)blockIdx.x * 256 + threadIdx.x;
  if (i >= TTOT) return;
  const int b = (int)(i / (TC * THW));
  const int hw = (int)(i % THW);
  const float cmin = cmm[b * 2], cmax = cmm[b * 2 + 1];
  const float cn = (cor[b * THW + hw] - cmin) / (cmax - cmin + 1e-12f);
  float v = Xc[i] * cn;
  if (addprev) v += Xcur[i];
  Xcur[i] = v;
}

// ---------------------------------------------------------------------------
// consen[c] = mean over (b,hw). grid 768, block 256
// ---------------------------------------------------------------------------
__global__ __launch_bounds__(256) void k_consen(const float* __restrict__ X,
                                                float* __restrict__ consen) {
  const int c = blockIdx.x;
  float s = 0.f;
  for (int idx = threadIdx.x; idx < TN; idx += 256) {
    const int b = idx / THW, hw = idx % THW;
    s += X[(size_t)b * (TC * THW) + (size_t)c * THW + hw];
  }
  __shared__ float red[256];
  red[threadIdx.x] = s;
  __syncthreads();
  for (int off = 128; off >= 1; off >>= 1) {
    if (threadIdx.x < off) red[threadIdx.x] += red[threadIdx.x + off];
    __syncthreads();
  }
  if (threadIdx.x == 0) consen[c] = red[0] / (float)TN;
}

__global__ __launch_bounds__(256) void k_final(const float* __restrict__ Xc,
                                               const float* __restrict__ Xin,
                                               const float* __restrict__ consen,
                                               float* __restrict__ out) {
  const size_t i = (size_t)blockIdx.x * 256 + threadIdx.x;
  if (i >= TTOT) return;
  const int c = (int)((i / THW) % TC);
  out[i] = Xc[i] + Xin[i] * consen[c];
}

// ---------------------------------------------------------------------------
extern "C" void kernel_launch(void* const* d_in, const int* in_sizes, int n_in,
                              void* d_out, int out_size, void* d_ws, size_t ws_size,
                              hipStream_t stream) {
  const float* x5_in   = (const float*)d_in[0];
  const float* conv_w  = (const float*)d_in[1];
  const float* conv_b  = (const float*)d_in[2];
  const float* query_w = (const float*)d_in[3];
  const float* query_b = (const float*)d_in[4];
  const float* key_w   = (const float*)d_in[5];
  const float* key_b   = (const float*)d_in[6];
  float* out = (float*)d_out;

  float* ws = (float*)d_ws;
  float* x_cur  = ws;
  float* x_conv = ws + TTOT;
  float* qbuf   = ws + 2 * TTOT;
  float* kbuf   = ws + 3 * TTOT;
  float* red2   = ws + 4 * TTOT;           // N*16
  float* maskb  = red2 + (size_t)TN * TB;  // N
  float* inorm  = maskb + TN;              // N
  float* seeds  = inorm + TN;              // 16*768
  float* cor    = seeds + TB * TC;         // N
  float* cmm    = cor + TN;                // 32
  float* consen = cmm + 2 * TB;            // 768

  const dim3 gGemm(9, 6, 16);
  const int gElem = (int)(TTOT / 256);  // 27648, exact

  for (int l = 0; l < 4; ++l) {
    const float* xin = (l == 0) ? x5_in : x_cur;
    const size_t wOff = (size_t)l * TC * TC;
    const size_t bOff = (size_t)l * TC;

    k_conv1x1<true><<<gGemm, 128, 0, stream>>>(conv_w + wOff, conv_b + bOff, xin, x_conv);
    k_conv1x1<false><<<gGemm, 128, 0, stream>>>(query_w + wOff, query_b + bOff, x_conv, qbuf);
    k_conv1x1<false><<<gGemm, 128, 0, stream>>>(key_w + wOff, key_b + bOff, x_conv, kbuf);

    k_attn_max<<<dim3(576, 16), 128, 0, stream>>>(qbuf, kbuf, red2);
    k_invnorm<<<36, 256, 0, stream>>>(x_conv, inorm);
    k_softmax_mask<<<16, 576, 0, stream>>>(red2, maskb);
    k_seeds<<<dim3(3, 16), 256, 0, stream>>>(x_conv, inorm, maskb, seeds);
    k_cor<<<36, 256, 0, stream>>>(x_conv, inorm, seeds, cor);
    k_minmax<<<16, 576, 0, stream>>>(cor, cmm);
    k_scale_res<<<gElem, 256, 0, stream>>>(x_conv, cor, cmm, x_cur, (l > 0) ? 1 : 0);
  }

  k_consen<<<TC, 256, 0, stream>>>(x_cur, consen);
  k_final<<<gElem, 256, 0, stream>>>(x_cur, x5_in, consen, out);
}